// GraphConvNet_51754355916837
// MI455X (gfx1250) — compile-verified
//
#include <hip/hip_runtime.h>
#include <math.h>

// CDNA5 / gfx1250: wave32, WMMA f32 16x16x4 for fp32 GEMM paths.
typedef __attribute__((ext_vector_type(2))) float v2f;
typedef __attribute__((ext_vector_type(8))) float v8f;

#define D 128
#define TILE_M 16
#define N_CLASSES 10
#define POOL_STRIP 64

// ---------------------------------------------------------------- zero fill
__global__ void gcn_zero_f4(float4* __restrict__ p, int n4) {
    int i = blockIdx.x * blockDim.x + threadIdx.x;
    if (i < n4) p[i] = make_float4(0.f, 0.f, 0.f, 0.f);
}

__global__ void gcn_zero_i32(int* __restrict__ p, int n) {
    int i = blockIdx.x * blockDim.x + threadIdx.x;
    if (i < n) p[i] = 0;
}

__global__ void gcn_copy_i32(const int* __restrict__ s, int* __restrict__ d, int n) {
    int i = blockIdx.x * blockDim.x + threadIdx.x;
    if (i < n) d[i] = s[i];
}

// ------------------------------------------------------------ CSR build
// Histogram of in-degrees (int atomics: 1.6M, one-time per call).
__global__ void gcn_hist(const int* __restrict__ dst, int* __restrict__ deg, int nEdges) {
    int e = blockIdx.x * blockDim.x + threadIdx.x;
    if (e < nEdges) atomicAdd(&deg[dst[e]], 1);
}

// Single-block exclusive prefix sum (Hillis-Steele over 1024-wide chunks with carry).
__global__ void __launch_bounds__(1024)
gcn_scan(const int* __restrict__ deg, int* __restrict__ off, int n) {
    __shared__ int sh[1024];
    __shared__ int carry_s;
    if (threadIdx.x == 0) carry_s = 0;
    __syncthreads();
    for (int base = 0; base < n; base += 1024) {
        int i = base + (int)threadIdx.x;
        int v = (i < n) ? deg[i] : 0;
        sh[threadIdx.x] = v;
        __syncthreads();
        for (int s = 1; s < 1024; s <<= 1) {
            int t = (threadIdx.x >= (unsigned)s) ? sh[threadIdx.x - s] : 0;
            __syncthreads();
            sh[threadIdx.x] += t;
            __syncthreads();
        }
        int c = carry_s;                       // everyone reads old carry
        if (i < n) off[i] = c + sh[threadIdx.x] - v;  // exclusive
        __syncthreads();
        if (threadIdx.x == 1023) carry_s = c + sh[1023];
        __syncthreads();
    }
    if (threadIdx.x == 0) off[n] = carry_s;
}

// Slot assignment: eidx[off[dst]+k] = src  (int atomics: 1.6M, one-time).
__global__ void gcn_fill(const int* __restrict__ src, const int* __restrict__ dst,
                         int* __restrict__ cursor, int* __restrict__ eidx, int nEdges) {
    int e = blockIdx.x * blockDim.x + threadIdx.x;
    if (e >= nEdges) return;
    int pos = atomicAdd(&cursor[dst[e]], 1);
    eidx[pos] = src[e];
}

// -------------------------------------------- gather-based aggregation
// One wave per node; lane owns 4 features. Pure L2-hit loads, no atomics,
// and no agg zero-fill pass. Replaces 204.8M f32 atomics per layer.
__global__ void gcn_aggregate(const float* __restrict__ x,
                              const int* __restrict__ off,
                              const int* __restrict__ eidx,
                              float* __restrict__ agg, int nNodes) {
    int t = blockIdx.x * blockDim.x + threadIdx.x;
    int node = t >> 5;
    if (node >= nNodes) return;
    int lane = t & 31;
    int beg = off[node], end = off[node + 1];
    float ax = 0.f, ay = 0.f, az = 0.f, aw = 0.f;
    for (int e = beg; e < end; ++e) {
        int s = eidx[e];  // uniform across the wave -> broadcast load
        const float4 v = *(const float4*)(x + (size_t)s * D + lane * 4);
        ax += v.x; ay += v.y; az += v.z; aw += v.w;
    }
    *(float4*)(agg + (size_t)node * D + lane * 4) = make_float4(ax, ay, az, aw);
}

// ------------------------------------------------------------- layer GEMM
// out = agg @ Wrel + brel + xin @ Wroot     (all fp32, WMMA 16x16x4)
// Block = 256 threads = 8 waves; wave w owns column tile [16w, 16w+16).
// Wrel/Wroot/brel preloaded to LDS (128.5 KB of the 320 KB WGP LDS),
// amortized across many 16-row M tiles via a grid-stride loop.
__global__ void __launch_bounds__(256)
gcn_layer_gemm(const float* __restrict__ agg, const float* __restrict__ xin,
               const float* __restrict__ Wrel, const float* __restrict__ Wroot,
               const float* __restrict__ brel, float* __restrict__ out,
               int nNodes) {
    __shared__ float lds[2 * D * D + D];
    float* Wr = lds;              // [128][128] row-major (k, n)
    float* Wo = lds + D * D;
    float* bs = lds + 2 * D * D;  // [128]

    for (int i = threadIdx.x; i < (D * D) / 4; i += blockDim.x) {
        ((float4*)Wr)[i] = ((const float4*)Wrel)[i];
        ((float4*)Wo)[i] = ((const float4*)Wroot)[i];
    }
    if (threadIdx.x < D / 4)
        ((float4*)bs)[threadIdx.x] = ((const float4*)brel)[threadIdx.x];
    __syncthreads();

    const int lane    = threadIdx.x & 31;
    const int wave    = threadIdx.x >> 5;   // 0..7 -> N tile
    const int nloc    = lane & 15;          // M index (A) / N index (B,C,D)
    const int khalf   = lane >> 4;          // 0: K=k..k+1, 1: K=k+2..k+3
    const int colBase = wave * 16;
    const int nTiles  = nNodes / TILE_M;    // nNodes % 16 == 0 for this model

    for (int mt = blockIdx.x; mt < nTiles; mt += gridDim.x) {
        const int rowBase = mt * TILE_M;
        const float* aRow = agg + (size_t)(rowBase + nloc) * D;
        const float* xRow = xin + (size_t)(rowBase + nloc) * D;

        // C/D layout: lanes 0-15 -> N=lane, M=r; lanes 16-31 -> N=lane-16, M=r+8
        const float bias = bs[colBase + nloc];
        v8f acc;
#pragma unroll
        for (int r = 0; r < 8; ++r) acc[r] = bias;

#pragma unroll 8
        for (int k = 0; k < D; k += 4) {
            const int kk = k + 2 * khalf;
            v2f a0 = *(const v2f*)(aRow + kk);
            v2f b0;
            b0.x = Wr[kk * D + colBase + nloc];
            b0.y = Wr[(kk + 1) * D + colBase + nloc];
            acc = __builtin_amdgcn_wmma_f32_16x16x4_f32(
                false, a0, false, b0, (short)0, acc, false, false);

            v2f a1 = *(const v2f*)(xRow + kk);
            v2f b1;
            b1.x = Wo[kk * D + colBase + nloc];
            b1.y = Wo[(kk + 1) * D + colBase + nloc];
            acc = __builtin_amdgcn_wmma_f32_16x16x4_f32(
                false, a1, false, b1, (short)0, acc, false, false);
        }

#pragma unroll
        for (int r = 0; r < 8; ++r) {
            int row = rowBase + r + 8 * khalf;
            out[(size_t)row * D + colBase + nloc] = acc[r];
        }
    }
}

// ---------------------------------------------- mean pool (sorted batch)
// Wave accumulates a contiguous strip of POOL_STRIP nodes in registers and
// flushes atomics only at graph boundaries (batch is sorted) -> ~200K atomics.
__global__ void gcn_pool(const float* __restrict__ x,
                         const int* __restrict__ batch,
                         float* __restrict__ pooled,
                         float* __restrict__ counts, int nNodes) {
    int wid = (blockIdx.x * blockDim.x + threadIdx.x) >> 5;
    int lane = threadIdx.x & 31;
    int beg = wid * POOL_STRIP;
    if (beg >= nNodes) return;
    int end = min(beg + POOL_STRIP, nNodes);

    float ax = 0.f, ay = 0.f, az = 0.f, aw = 0.f;
    int cnt = 0;
    int curg = batch[beg];
    for (int n = beg; n < end; ++n) {
        int g = batch[n];  // uniform across wave
        if (g != curg) {
            float* p = pooled + (size_t)curg * D + lane * 4;
            atomicAdd(p + 0, ax); atomicAdd(p + 1, ay);
            atomicAdd(p + 2, az); atomicAdd(p + 3, aw);
            if (lane == 0) atomicAdd(counts + curg, (float)cnt);
            ax = ay = az = aw = 0.f; cnt = 0; curg = g;
        }
        const float4 v = *(const float4*)(x + (size_t)n * D + lane * 4);
        ax += v.x; ay += v.y; az += v.z; aw += v.w;
        ++cnt;
    }
    float* p = pooled + (size_t)curg * D + lane * 4;
    atomicAdd(p + 0, ax); atomicAdd(p + 1, ay);
    atomicAdd(p + 2, az); atomicAdd(p + 3, aw);
    if (lane == 0) atomicAdd(counts + curg, (float)cnt);
}

// ------------------------------------------------- FC head + log_softmax
__global__ void gcn_head(const float* __restrict__ pooled,
                         const float* __restrict__ counts,
                         const float* __restrict__ Wfc,
                         const float* __restrict__ bfc,
                         float* __restrict__ out, int nGraphs) {
    int g = blockIdx.x * blockDim.x + threadIdx.x;
    if (g >= nGraphs) return;
    float inv = 1.0f / fmaxf(counts[g], 1.0f);
    float logit[N_CLASSES];
#pragma unroll
    for (int c = 0; c < N_CLASSES; ++c) logit[c] = bfc[c];
    for (int d = 0; d < D; ++d) {
        float pv = pooled[(size_t)g * D + d] * inv;
#pragma unroll
        for (int c = 0; c < N_CLASSES; ++c) logit[c] += pv * Wfc[d * N_CLASSES + c];
    }
    float m = logit[0];
#pragma unroll
    for (int c = 1; c < N_CLASSES; ++c) m = fmaxf(m, logit[c]);
    float s = 0.f;
#pragma unroll
    for (int c = 0; c < N_CLASSES; ++c) s += expf(logit[c] - m);
    float lse = logf(s);
#pragma unroll
    for (int c = 0; c < N_CLASSES; ++c) out[g * N_CLASSES + c] = logit[c] - m - lse;
}

extern "C" void kernel_launch(void* const* d_in, const int* in_sizes, int n_in,
                              void* d_out, int out_size, void* d_ws, size_t ws_size,
                              hipStream_t stream) {
    const float* x     = (const float*)d_in[0];
    const int*   edge  = (const int*)d_in[1];   // [2, E]: row0=src, row1=dst
    const int*   batch = (const int*)d_in[2];
    const float* Wrel  = (const float*)d_in[3]; // [3,128,128]
    const float* Wroot = (const float*)d_in[4]; // [3,128,128]
    const float* brel  = (const float*)d_in[5]; // [3,128]
    const float* Wfc   = (const float*)d_in[6]; // [128,10]
    const float* bfc   = (const float*)d_in[7]; // [10]

    const int nNodes  = in_sizes[0] / D;
    const int nEdges  = in_sizes[1] / 2;
    const int nGraphs = out_size / N_CLASSES;
    const int* src = edge;
    const int* dst = edge + nEdges;

    // workspace: agg | buf0 | buf1 | pooled | counts | deg | off | cursor | eidx
    float* ws = (float*)d_ws;
    size_t nd = (size_t)nNodes * D;
    float* agg    = ws;
    float* buf0   = ws + nd;
    float* buf1   = ws + 2 * nd;
    float* pooled = ws + 3 * nd;
    float* counts = pooled + (size_t)nGraphs * D;
    int*   deg    = (int*)(counts + nGraphs);
    int*   off    = deg + nNodes;          // nNodes+1 entries
    int*   cursor = off + nNodes + 1;
    int*   eidx   = cursor + nNodes;       // nEdges entries

    const int TB = 256;

    // ---- CSR-by-destination build (once per call, deterministic) ----
    gcn_zero_i32<<<(nNodes + TB - 1) / TB, TB, 0, stream>>>(deg, nNodes);
    gcn_hist<<<(nEdges + TB - 1) / TB, TB, 0, stream>>>(dst, deg, nEdges);
    gcn_scan<<<1, 1024, 0, stream>>>(deg, off, nNodes);
    gcn_copy_i32<<<(nNodes + TB - 1) / TB, TB, 0, stream>>>(off, cursor, nNodes);
    gcn_fill<<<(nEdges + TB - 1) / TB, TB, 0, stream>>>(src, dst, cursor, eidx, nEdges);

    // ---- 3 GraphConv layers ----
    const float* cur = x;
    float* nxt = buf0;
    for (int layer = 0; layer < 3; ++layer) {
        long tN = (long)nNodes * 32;
        gcn_aggregate<<<(int)((tN + TB - 1) / TB), TB, 0, stream>>>(
            cur, off, eidx, agg, nNodes);
        gcn_layer_gemm<<<512, 256, 0, stream>>>(
            agg, cur, Wrel + (size_t)layer * D * D, Wroot + (size_t)layer * D * D,
            brel + (size_t)layer * D, nxt, nNodes);
        cur = nxt;
        nxt = (layer == 0) ? buf1 : buf0;
    }

    // ---- mean pool + head ----
    int pz4 = (nGraphs * D + nGraphs + 3) / 4;  // pooled + counts contiguous
    gcn_zero_f4<<<(pz4 + TB - 1) / TB, TB, 0, stream>>>((float4*)pooled, pz4);

    int nWaves = (nNodes + POOL_STRIP - 1) / POOL_STRIP;
    long tP = (long)nWaves * 32;
    gcn_pool<<<(int)((tP + TB - 1) / TB), TB, 0, stream>>>(cur, batch, pooled, counts, nNodes);

    gcn_head<<<1, 64, 0, stream>>>(pooled, counts, Wfc, bfc, (float*)d_out, nGraphs);
}